// FCOSAnchor_82248623718462
// MI455X (gfx1250) — compile-verified
//
#include <hip/hip_runtime.h>

#define INF_VAL     100000000.0f
#define NUM_CLASSES 80
#define GBOX        64

// FCOS anchor matching for MI455X (gfx1250).
// CDNA5-specific path: per-block async global->LDS staging of the GT box
// table (ASYNCcnt + s_wait_asynccnt), then wave32 LDS-broadcast inner loop.
__global__ __launch_bounds__(256) void fcos_match_kernel(
    const float* __restrict__ locations,       // (L,2)
    const float* __restrict__ strides_per_loc, // (L,)
    const float* __restrict__ soi,             // (L,2)
    const float* __restrict__ anchors,         // (L,4)
    const float* __restrict__ gt_boxes,        // (B,64,4)
    const int*   __restrict__ gt_classes,      // (B,64)
    float* __restrict__ out,                   // [B*L cls][B*L*4 reg][B*L*4 deltas]
    int L, int B)
{
    __shared__ __align__(16) float  sb[GBOX * 4]; // raw boxes x0,y0,x1,y1
    __shared__ __align__(16) float4 sd[GBOX];     // ccx, ccy, area, pad
    __shared__ __align__(16) int    sc[GBOX];     // classes

    const int tid = threadIdx.x;
    const int b   = blockIdx.y;

    // ---- async stage: 64 lanes pull 16B of boxes each, 16 lanes pull classes ----
    if (tid < GBOX) {
        unsigned lds = (unsigned)(unsigned long long)(void*)&sb[tid * 4];
        unsigned long long g =
            (unsigned long long)(gt_boxes + ((size_t)b * GBOX + (size_t)tid) * 4);
        asm volatile("global_load_async_to_lds_b128 %0, %1, off"
                     :: "v"(lds), "v"(g) : "memory");
    } else if (tid < GBOX + 16) {
        int k = tid - GBOX;
        unsigned lds = (unsigned)(unsigned long long)(void*)&sc[k * 4];
        unsigned long long g =
            (unsigned long long)(gt_classes + (size_t)b * GBOX + (size_t)k * 4);
        asm volatile("global_load_async_to_lds_b128 %0, %1, off"
                     :: "v"(lds), "v"(g) : "memory");
    }
    asm volatile("s_wait_asynccnt 0x0" ::: "memory");
    __syncthreads();

    // ---- precompute per-box center + area ----
    if (tid < GBOX) {
        float x0 = sb[4 * tid + 0], y0 = sb[4 * tid + 1];
        float x1 = sb[4 * tid + 2], y1 = sb[4 * tid + 3];
        sd[tid] = make_float4(0.5f * (x0 + x1), 0.5f * (y0 + y1),
                              (x1 - x0) * (y1 - y0), 0.0f);
    }
    __syncthreads();

    const int i = blockIdx.x * blockDim.x + tid;
    if (i >= L) return;

    const float2 xy  = ((const float2*)locations)[i];
    const float  x   = xy.x, y = xy.y;
    const float  rad = strides_per_loc[i] * 1.5f;
    const float2 sz  = ((const float2*)soi)[i];
    const float  lo  = sz.x, hi = sz.y;

    const float4* sb4 = (const float4*)sb;

    float best_area = INF_VAL;
    int   best      = 0;
#pragma unroll 8
    for (int j = 0; j < GBOX; ++j) {
        float4 bb = sb4[j];           // x0,y0,x1,y1  (LDS broadcast)
        float4 dd = sd[j];            // ccx,ccy,area
        float l  = x - bb.x, t = y - bb.y, r = bb.z - x, bt = bb.w - y;
        float mx = fmaxf(fmaxf(l, t), fmaxf(r, bt));   // max_reg
        float mn = fminf(fminf(l, t), fminf(r, bt));
        float dc = fmaxf(fabsf(x - dd.x), fabsf(y - dd.y));
        // in_box(center-sampled) & cared
        bool ok  = (mn > 0.0f) && (dc < rad) && (mx >= lo) && (mx <= hi);
        float val = ok ? dd.z : INF_VAL;
        if (val < best_area) { best_area = val; best = j; }  // first-argmin
    }

    const float4 bb = sb4[best];
    const float cls = (best_area == INF_VAL) ? (float)NUM_CLASSES
                                             : (float)sc[best];

    const float l  = x - bb.x, t = y - bb.y, r = bb.z - x, bt = bb.w - y;
    const float gw = bb.z - bb.x, gh = bb.w - bb.y;
    const float gcx = bb.x + 0.5f * gw, gcy = bb.y + 0.5f * gh;

    const float4 a  = ((const float4*)anchors)[i];
    const float aw  = a.z - a.x, ah = a.w - a.y;
    const float acx = a.x + 0.5f * aw, acy = a.y + 0.5f * ah;

    const size_t BL  = (size_t)B * (size_t)L;
    const size_t idx = (size_t)b * (size_t)L + (size_t)i;

    out[idx] = cls;
    float4* out_reg = (float4*)(out + BL);
    float4* out_del = (float4*)(out + 5 * BL);
    out_reg[idx] = make_float4(l, t, r, bt);
    out_del[idx] = make_float4((gcx - acx) / aw, (gcy - acy) / ah,
                               logf(gw / aw),    logf(gh / ah));
}

extern "C" void kernel_launch(void* const* d_in, const int* in_sizes, int n_in,
                              void* d_out, int out_size, void* d_ws, size_t ws_size,
                              hipStream_t stream) {
    const float* locations = (const float*)d_in[0];
    const float* strides   = (const float*)d_in[1];
    const float* soi       = (const float*)d_in[2];
    const float* anchors   = (const float*)d_in[3];
    const float* gt_boxes  = (const float*)d_in[4];
    const int*   gt_cls    = (const int*)d_in[5];

    const int L = in_sizes[1];          // strides_per_loc has L elements
    const int B = in_sizes[5] / GBOX;   // gt_classes has B*64 elements

    dim3 grid((L + 255) / 256, B, 1);
    fcos_match_kernel<<<grid, 256, 0, stream>>>(
        locations, strides, soi, anchors, gt_boxes, gt_cls,
        (float*)d_out, L, B);
}